// PieceWisePlanarRegularization_27857157882375
// MI455X (gfx1250) — compile-verified
//
#include <hip/hip_runtime.h>
#include <hip/hip_bf16.h>
#include <math.h>

// ---- problem constants (match reference) ----
#define HH 1024
#define WW 1024
#define NN (HH * WW)          // 1,048,576 pixels
#define KK 15
#define GAMMA_C 0.5f
#define MULT_C 1.0f

#define BLOCK 256
#define PPT 4                 // pixels per thread (float4-wide streams)
#define NBLK (NN / (BLOCK * PPT))   // 1024 blocks

typedef __attribute__((ext_vector_type(2))) float v2f;
typedef __attribute__((ext_vector_type(4))) float v4f;
typedef __attribute__((ext_vector_type(8))) float v8f;
typedef __attribute__((ext_vector_type(4))) int   v4i;

// Cross-lane sum of 32 per-lane f32 values using V_WMMA_F32_16X16X4_F32.
// A = ones(16x4), B holds each lane's value exactly once (4x16) ->
// D[m][n] = colsum(B)[n] for every m; row 0 lives in c[0] of lanes 0..15.
// Valid result in lane 0.
__device__ __forceinline__ float wave_reduce_wmma(float v) {
    v2f a; a[0] = 1.0f; a[1] = 1.0f;
    v2f b; b[0] = v;    b[1] = 0.0f;
    v8f c = {};
    c = __builtin_amdgcn_wmma_f32_16x16x4_f32(
        /*neg_a=*/false, a, /*neg_b=*/false, b,
        /*c_mod=*/(short)0, c, /*reuse_a=*/false, /*reuse_b=*/false);
    float s = c[0];                 // lanes 0..15 hold the 16 column sums
    s += __shfl_down(s, 8, 32);
    s += __shfl_down(s, 4, 32);
    s += __shfl_down(s, 2, 32);
    s += __shfl_down(s, 1, 32);
    return s;                       // lane 0: sum of all 32 inputs
}

// Block reduction (wave32: 8 waves per 256-thread block). Result on thread 0.
__device__ __forceinline__ float block_reduce(float v) {
    __shared__ float sm[BLOCK / 32];
    const int lane = threadIdx.x & 31;
    const int wid  = threadIdx.x >> 5;
    float ws = wave_reduce_wmma(v);
    if (lane == 0) sm[wid] = ws;
    __syncthreads();
    float t = 0.0f;
    if (threadIdx.x == 0) {
#pragma unroll
        for (int i = 0; i < BLOCK / 32; ++i) t += sm[i];
    }
    return t;
}

__global__ __launch_bounds__(BLOCK) void pwpr_main(
    const float* __restrict__ s1,     // (N,)
    const float* __restrict__ s2,     // (2, N)
    const float* __restrict__ wgt,    // (K, N)
    const float* __restrict__ dist,   // (K, 2, N)
    const int*   __restrict__ nbr,    // (K, N)
    float* __restrict__ blockPart)    // (NBLK,)
{
    const int base = (blockIdx.x * BLOCK + threadIdx.x) * PPT;

    // Self values: regular (cached) loads — these arrays are also gathered.
    const v4f c1  = *(const v4f*)(s1 + base);
    const v4f c2x = *(const v4f*)(s2 + base);
    const v4f c2y = *(const v4f*)(s2 + NN + base);

    float sum1_0 = 0.0f, sum1_1 = 0.0f, sum1_2 = 0.0f, sum1_3 = 0.0f;
    float acc2 = 0.0f;

    for (int k = 0; k < KK; ++k) {
        // Zero-reuse streams: non-temporal b128 loads so they don't evict
        // the reusable s1/s2 gather window from cache.
        const v4f wk = __builtin_nontemporal_load(
            (const v4f*)(wgt + (size_t)k * NN + base));
        const v4f dx = __builtin_nontemporal_load(
            (const v4f*)(dist + (size_t)(2 * k) * NN + base));
        const v4f dy = __builtin_nontemporal_load(
            (const v4f*)(dist + (size_t)(2 * k + 1) * NN + base));
        const v4i jk = __builtin_nontemporal_load(
            (const v4i*)(nbr + (size_t)k * NN + base));

        float s1acc[4];
#pragma unroll
        for (int p = 0; p < 4; ++p) {
            const int j = jk[p];
            const float s1j  = s1[j];
            const float s2xj = s2[j];
            const float s2yj = s2[NN + j];

            const float plane = fmaf(c2x[p], dx[p], c2y[p] * dy[p]);
            const float a1 = (c1[p] - s1j - plane) * wk[p];
            s1acc[p] = a1 * a1;

            const float ex = c2x[p] - s2xj;
            const float ey = c2y[p] - s2yj;
            acc2 = fmaf(__builtin_sqrtf(fmaf(ex, ex, ey * ey)), wk[p], acc2);
        }
        sum1_0 += s1acc[0];
        sum1_1 += s1acc[1];
        sum1_2 += s1acc[2];
        sum1_3 += s1acc[3];
    }

    const float part = __builtin_sqrtf(sum1_0) + __builtin_sqrtf(sum1_1) +
                       __builtin_sqrtf(sum1_2) + __builtin_sqrtf(sum1_3) +
                       GAMMA_C * acc2;

    const float bsum = block_reduce(part);
    if (threadIdx.x == 0) blockPart[blockIdx.x] = bsum;
}

__global__ __launch_bounds__(BLOCK) void pwpr_finish(
    const float* __restrict__ blockPart, float* __restrict__ out, int n)
{
    float v = 0.0f;
    for (int i = threadIdx.x; i < n; i += BLOCK) v += blockPart[i];
    const float total = block_reduce(v);
    if (threadIdx.x == 0)
        out[0] = MULT_C * total * (1.0f / (float)NN);
}

extern "C" void kernel_launch(void* const* d_in, const int* in_sizes, int n_in,
                              void* d_out, int out_size, void* d_ws, size_t ws_size,
                              hipStream_t stream) {
    (void)in_sizes; (void)n_in; (void)out_size; (void)ws_size;
    const float* s1   = (const float*)d_in[0];   // sig1 (1,1,H,W)
    const float* s2   = (const float*)d_in[1];   // sig2 (1,2,H,W)
    const float* wgt  = (const float*)d_in[2];   // weights (K,N)
    const float* dist = (const float*)d_in[3];   // dist (K,2,N)
    const int*   nbr  = (const int*)d_in[4];     // neighbours (K,N)
    float* partials = (float*)d_ws;              // NBLK floats of scratch

    pwpr_main<<<NBLK, BLOCK, 0, stream>>>(s1, s2, wgt, dist, nbr, partials);
    pwpr_finish<<<1, BLOCK, 0, stream>>>(partials, (float*)d_out, NBLK);
}